// PointNet2_12034498363930
// MI455X (gfx1250) — compile-verified
//
#include <hip/hip_runtime.h>
#include <cstdint>
#include <cstddef>

// ============================================================================
// PointNet++ (SSG) encoder for MI455X / gfx1250, f32 end-to-end.
// GEMMs use the native-f32 CDNA5 tensor op V_WMMA_F32_16X16X4_F32.
// K-dim is padded to a multiple of 4 so the WMMA inner loop is branch-free:
// two global_load_b64 + one v_wmma per step, dual accumulators to hide the
// C-operand RAW latency. wave32 everywhere.
// ============================================================================

typedef __attribute__((ext_vector_type(2))) float v2f;
typedef __attribute__((ext_vector_type(8))) float v8f;

#define BATCH 16

// ---------------- transpose (B,3,N) -> (B,N,3) -----------------------------
__global__ void k_transpose(const float* __restrict__ in, float* __restrict__ out, int N) {
  int t = blockIdx.x * blockDim.x + threadIdx.x;
  int total = BATCH * N * 3;
  if (t >= total) return;
  int c = t % 3; int r = t / 3; int n = r % N; int b = r / N;
  out[t] = in[((size_t)b * 3 + c) * N + n];
}

// ---------------- farthest point sampling ----------------------------------
// One workgroup per batch; cloud staged in LDS (48KB of the 320KB WGP pool),
// running min-distance in registers. Tie-break matches jnp.argmax (smallest
// index among equal maxima).
__global__ __launch_bounds__(256) void k_fps(const float* __restrict__ xyz,
                                             int* __restrict__ out, int N, int npoint) {
  __shared__ float sx[4096], sy[4096], sz[4096];
  __shared__ float rv[256];
  __shared__ int   ri[256];
  __shared__ int   scur;
  int b = blockIdx.x, tid = threadIdx.x;
  const float* base = xyz + (size_t)b * N * 3;
  for (int i = tid; i < N; i += 256) { sx[i] = base[i*3]; sy[i] = base[i*3+1]; sz[i] = base[i*3+2]; }
  float dist[16];
#pragma unroll
  for (int j = 0; j < 16; ++j) dist[j] = 1e10f;
  if (tid == 0) scur = 0;
  __syncthreads();
  for (int it = 0; it < npoint; ++it) {
    int cur = scur;
    if (tid == 0) out[b * npoint + it] = cur;
    float cx = sx[cur], cy = sy[cur], cz = sz[cur];
    float lmax = -1.0f; int li = 0;
#pragma unroll
    for (int j = 0; j < 16; ++j) {
      int i = tid + j * 256;
      if (i < N) {
        float dx = sx[i]-cx, dy = sy[i]-cy, dz = sz[i]-cz;
        float d  = dx*dx + dy*dy + dz*dz;
        float nd = fminf(dist[j], d);
        dist[j] = nd;
        if (nd > lmax) { lmax = nd; li = i; }
      }
    }
    rv[tid] = lmax; ri[tid] = li;
    __syncthreads();
    for (int s = 128; s > 0; s >>= 1) {
      if (tid < s) {
        float v2 = rv[tid + s]; int i2 = ri[tid + s];
        if (v2 > rv[tid] || (v2 == rv[tid] && i2 < ri[tid])) { rv[tid] = v2; ri[tid] = i2; }
      }
      __syncthreads();
    }
    if (tid == 0) scur = ri[0];
    __syncthreads();
  }
}

// ---------------- gather sampled centers -----------------------------------
__global__ void k_gather_nx(const float* __restrict__ pts, const int* __restrict__ fidx,
                            float* __restrict__ out, int N, int S) {
  int t = blockIdx.x * blockDim.x + threadIdx.x;
  if (t >= BATCH * S * 3) return;
  int c = t % 3; int r = t / 3; int s = r % S; int b = r / S;
  int p = fidx[b * S + s];
  out[t] = pts[((size_t)b * N + p) * 3 + c];
}

// ---------------- ball query (wave32 ordered compaction) -------------------
__global__ __launch_bounds__(256) void k_ball_query(const float* __restrict__ xyz,
                                                    const float* __restrict__ nxyz,
                                                    int* __restrict__ idx,
                                                    int N, int S, int K, float r2) {
  int w = blockIdx.x * (blockDim.x >> 5) + (threadIdx.x >> 5);
  if (w >= BATCH * S) return;          // wave-uniform
  int lane = threadIdx.x & 31;
  int b = w / S, s = w % S;
  const float* c = nxyz + ((size_t)b * S + s) * 3;
  float cx = c[0], cy = c[1], cz = c[2];
  int* od = idx + (size_t)w * K;
  int count = 0;
  int firstN = -1;
  for (int n0 = 0; n0 < N && count < K; n0 += 32) {
    int n = n0 + lane;
    bool ok = false;
    if (n < N) {
      const float* p = xyz + ((size_t)b * N + n) * 3;
      float dx = p[0]-cx, dy = p[1]-cy, dz = p[2]-cz;
      ok = (dx*dx + dy*dy + dz*dz) <= r2;
    }
    unsigned mask = (unsigned)__ballot(ok);
    int pos = count + __popc(mask & ((1u << lane) - 1u));
    if (ok && pos < K)  od[pos] = n;
    if (ok && pos == 0) firstN = n;
    count += __popc(mask);
  }
  if (count > K) count = K;
  unsigned hf = (unsigned)__ballot(firstN >= 0);
  int fsrc = hf ? (__ffs((int)hf) - 1) : 0;
  int f = __shfl(firstN, fsrc, 32);
  if (f < 0) f = 0;
  for (int j = count + lane; j < K; j += 32) od[j] = f;
}

// ---------------- grouped feature build, padded channel stride -------------
// out rows are (B*S*K) x cinp; channels [cin,cinp) written as zero.
__global__ void k_build_features(const float* __restrict__ xyz, const float* __restrict__ pts,
                                 const float* __restrict__ nxyz, const int* __restrict__ idx,
                                 float* __restrict__ out, int N, int S, int K, int Cp, int cinp) {
  int cin = 3 + Cp;
  size_t t = (size_t)blockIdx.x * blockDim.x + threadIdx.x;
  size_t total = (size_t)BATCH * S * K * cinp;
  if (t >= total) return;
  int c = (int)(t % cinp); size_t row = t / cinp;
  float v = 0.0f;
  if (c < cin) {
    int k = (int)(row % K); size_t r1 = row / K;
    int s = (int)(r1 % S); int b = (int)(r1 / S);
    int p = idx[((size_t)b * S + s) * K + k];
    if (c < 3) v = xyz[((size_t)b * N + p) * 3 + c] - nxyz[((size_t)b * S + s) * 3 + c];
    else       v = pts[((size_t)b * N + p) * Cp + (c - 3)];
  }
  out[t] = v;
}

// stage-3 group_all: concat(l2_xyz, l2_p) -> rows (B*128) x 260 (259 + pad)
__global__ void k_build_concat3(const float* __restrict__ l2xyz, const float* __restrict__ l2p,
                                float* __restrict__ out) {
  const int cinp = 260;
  int t = blockIdx.x * blockDim.x + threadIdx.x;
  if (t >= 2048 * cinp) return;
  int c = t % cinp; int row = t / cinp;          // row = b*128 + n
  float v = 0.0f;
  if (c < 3)        v = l2xyz[row * 3 + c];
  else if (c < 259) v = l2p[(size_t)row * 256 + (c - 3)];
  out[t] = v;
}

// ---------------- zero-pad weights (co x cin) -> (co x cinp) ---------------
__global__ void k_pad_w(const float* __restrict__ W, float* __restrict__ Wp,
                        int co, int cin, int cinp) {
  int t = blockIdx.x * blockDim.x + threadIdx.x;
  if (t >= co * cinp) return;
  int c = t % cinp; int r = t / cinp;
  Wp[t] = (c < cin) ? W[(size_t)r * cin + c] : 0.0f;
}

// ---------------- GEMM via V_WMMA_F32_16X16X4_F32 --------------------------
// Y[M x co] = X[M x cinp(padded)] @ W[co x cinp]^T + bias.
// One wave = one 16x16 tile. cinp % 4 == 0 -> branch-free inner loop:
// aligned float2 (global_load_b64) A/B fragment loads, dual accumulators to
// break the C-operand RAW chain between consecutive WMMAs.
// ISA layouts: A lanes 0-15 hold row M=l with K={k,k+1}; lanes 16-31 the same
// rows with K={k+2,k+3}; B mirrors with N=l; D: VGPR r -> M=r+8*(lane>=16).
__global__ __launch_bounds__(256) void k_gemm_wmma(const float* __restrict__ X,
                                                   const float* __restrict__ Wm,
                                                   const float* __restrict__ bias,
                                                   float* __restrict__ Y,
                                                   int M, int cinp, int co) {
  int wave   = blockIdx.x * (blockDim.x >> 5) + (threadIdx.x >> 5);
  int tilesN = co >> 4;
  int tilesM = M >> 4;
  if (wave >= tilesM * tilesN) return;           // wave-uniform
  int tm   = (wave / tilesN) << 4;
  int tn   = (wave % tilesN) << 4;
  int lane = threadIdx.x & 31;
  int half = lane >> 4;                          // 0: K={k,k+1}, 1: K={k+2,k+3}
  int l    = lane & 15;
  const float* xr = X  + (size_t)(tm + l) * cinp + 2 * half;
  const float* wr = Wm + (size_t)(tn + l) * cinp + 2 * half;
  v8f acc0 = {}, acc1 = {};
  int k = 0;
  for (; k + 8 <= cinp; k += 8) {
    v2f a0 = *(const v2f*)(xr + k);
    v2f b0 = *(const v2f*)(wr + k);
    v2f a1 = *(const v2f*)(xr + k + 4);
    v2f b1 = *(const v2f*)(wr + k + 4);
    acc0 = __builtin_amdgcn_wmma_f32_16x16x4_f32(false, a0, false, b0, (short)0, acc0,
                                                 false, false);
    acc1 = __builtin_amdgcn_wmma_f32_16x16x4_f32(false, a1, false, b1, (short)0, acc1,
                                                 false, false);
  }
  if (k < cinp) {                                // wave-uniform tail (cinp%8==4)
    v2f a0 = *(const v2f*)(xr + k);
    v2f b0 = *(const v2f*)(wr + k);
    acc0 = __builtin_amdgcn_wmma_f32_16x16x4_f32(false, a0, false, b0, (short)0, acc0,
                                                 false, false);
  }
  float bb = bias[tn + l];
  int mb = tm + half * 8;
#pragma unroll
  for (int r = 0; r < 8; ++r)
    Y[(size_t)(mb + r) * co + (tn + l)] = acc0[r] + acc1[r] + bb;
}

// ---------------- batch-norm over all (B,S,K) rows -------------------------
__global__ void k_zero(float* __restrict__ p, int n) {
  int t = blockIdx.x * blockDim.x + threadIdx.x;
  if (t < n) p[t] = 0.0f;
}

__global__ __launch_bounds__(256) void k_bn_stats(const float* __restrict__ Y,
                                                  float* __restrict__ stats, int M, int co) {
  extern __shared__ float ls[];                  // [0,co)=sum, [co,2co)=sumsq
  unsigned cmask = (unsigned)co - 1u;            // co is a power of two
  for (int i = threadIdx.x; i < 2 * co; i += blockDim.x) ls[i] = 0.0f;
  __syncthreads();
  size_t total = (size_t)M * co;
  for (size_t i = (size_t)blockIdx.x * blockDim.x + threadIdx.x; i < total;
       i += (size_t)gridDim.x * blockDim.x) {
    float v = Y[i]; int c = (int)((unsigned)i & cmask);
    atomicAdd(&ls[c], v);
    atomicAdd(&ls[co + c], v * v);
  }
  __syncthreads();
  for (int i = threadIdx.x; i < 2 * co; i += blockDim.x) atomicAdd(&stats[i], ls[i]);
}

__global__ void k_bn_finalize(float* __restrict__ stats, const float* __restrict__ g,
                              const float* __restrict__ bt, int M, int co) {
  int c = blockIdx.x * blockDim.x + threadIdx.x;
  if (c >= co) return;
  float inv = 1.0f / (float)M;
  float mu  = stats[c] * inv;
  float var = stats[co + c] * inv - mu * mu;
  float sc  = g[c] * rsqrtf(var + 1e-5f);
  stats[2 * co + c] = sc;
  stats[3 * co + c] = bt[c] - mu * sc;
}

__global__ void k_bn_apply_relu(float* __restrict__ Y, const float* __restrict__ stats,
                                size_t total, int co) {
  size_t t = (size_t)blockIdx.x * blockDim.x + threadIdx.x;
  if (t >= total) return;
  int c = (int)((unsigned)t & ((unsigned)co - 1u));
  float v = Y[t] * stats[2 * co + c] + stats[3 * co + c];
  Y[t] = fmaxf(v, 0.0f);
}

// ---------------- max over neighborhood dimension K ------------------------
__global__ void k_max_k(const float* __restrict__ Y, float* __restrict__ out,
                        int BS, int K, int co) {
  int t = blockIdx.x * blockDim.x + threadIdx.x;
  if (t >= BS * co) return;
  int c = t & (co - 1); int bs = t / co;
  const float* p = Y + (size_t)bs * K * co + c;
  float m = p[0];
  for (int k = 1; k < K; ++k) m = fmaxf(m, p[(size_t)k * co]);
  out[t] = m;
}

// ============================================================================
extern "C" void kernel_launch(void* const* d_in, const int* in_sizes, int n_in,
                              void* d_out, int out_size, void* d_ws, size_t ws_size,
                              hipStream_t stream) {
  (void)in_sizes; (void)n_in; (void)out_size; (void)ws_size;
  const float* xyz_in = (const float*)d_in[0];
  // params flattened as 9 layers x (W, b, gamma, beta)
  auto Wp = [&](int j) { return (const float*)d_in[1 + 4 * j]; };
  auto Bp = [&](int j) { return (const float*)d_in[2 + 4 * j]; };
  auto Gp = [&](int j) { return (const float*)d_in[3 + 4 * j]; };
  auto Tp = [&](int j) { return (const float*)d_in[4 + 4 * j]; };

  char* ws = (char*)d_ws;
  size_t off = 0;
  auto allocF = [&](size_t n) { float* p = (float*)(ws + off); off += n * sizeof(float); return p; };
  auto allocI = [&](size_t n) { int*   p = (int*)(ws + off);   off += n * sizeof(int);   return p; };
  float* BUF0  = allocF(17301504);          // ping (max 131072 x 132)
  float* BUF1  = allocF(33554432);          // pong (max 131072 x 256)
  float* XT    = allocF((size_t)16 * 4096 * 3);
  float* NX1   = allocF((size_t)16 * 512 * 3);
  float* L1P   = allocF((size_t)16 * 512 * 128);
  float* NX2   = allocF((size_t)16 * 128 * 3);
  float* L2P   = allocF((size_t)16 * 128 * 256);
  float* STATS = allocF(4096);              // sum | sumsq | scale | shift
  float* WPAD  = allocF(524288);            // padded weights (max 1024x512)
  int*   FPS1  = allocI(16 * 512);
  int*   IDX1  = allocI((size_t)16 * 512 * 21);
  int*   FPS2  = allocI(16 * 128);
  int*   IDX2  = allocI((size_t)16 * 128 * 64);

  auto cdiv = [](size_t a, size_t b) { return (int)((a + b - 1) / b); };
  auto run_layer = [&](const float* Xb, float* Yb, int j, int M, int cin, int co) {
    int cinp = (cin + 3) & ~3;
    const float* Wuse = Wp(j);
    if (cinp != cin) {                      // pre-pad K-dim of the weights
      k_pad_w<<<cdiv((size_t)co * cinp, 256), 256, 0, stream>>>(Wp(j), WPAD, co, cin, cinp);
      Wuse = WPAD;
    }
    k_zero<<<cdiv(4 * (size_t)co, 256), 256, 0, stream>>>(STATS, 4 * co);
    int tiles = (M / 16) * (co / 16);
    k_gemm_wmma<<<cdiv((size_t)tiles, 8), 256, 0, stream>>>(Xb, Wuse, Bp(j), Yb, M, cinp, co);
    size_t total = (size_t)M * co;
    int sblocks = (int)((total + 255) / 256); if (sblocks > 2048) sblocks = 2048;
    k_bn_stats<<<sblocks, 256, 2 * co * sizeof(float), stream>>>(Yb, STATS, M, co);
    k_bn_finalize<<<cdiv((size_t)co, 256), 256, 0, stream>>>(STATS, Gp(j), Tp(j), M, co);
    k_bn_apply_relu<<<cdiv(total, 256), 256, 0, stream>>>(Yb, STATS, total, co);
  };

  // -------- stage 1: N=4096 -> S=512, K=21, r=0.2, MLP 3->64->64->128 ------
  k_transpose<<<cdiv((size_t)16 * 4096 * 3, 256), 256, 0, stream>>>(xyz_in, XT, 4096);
  k_fps<<<16, 256, 0, stream>>>(XT, FPS1, 4096, 512);
  k_gather_nx<<<cdiv((size_t)16 * 512 * 3, 256), 256, 0, stream>>>(XT, FPS1, NX1, 4096, 512);
  k_ball_query<<<cdiv((size_t)16 * 512, 8), 256, 0, stream>>>(XT, NX1, IDX1, 4096, 512, 21, 0.04f);
  k_build_features<<<cdiv((size_t)16 * 512 * 21 * 4, 256), 256, 0, stream>>>(
      XT, nullptr, NX1, IDX1, BUF0, 4096, 512, 21, 0, 4);
  const int M1 = 16 * 512 * 21;
  run_layer(BUF0, BUF1, 0, M1, 3, 64);
  run_layer(BUF1, BUF0, 1, M1, 64, 64);
  run_layer(BUF0, BUF1, 2, M1, 64, 128);
  k_max_k<<<cdiv((size_t)16 * 512 * 128, 256), 256, 0, stream>>>(BUF1, L1P, 16 * 512, 21, 128);

  // -------- stage 2: 512 -> 128, K=64, r=0.4, MLP 131->128->128->256 -------
  k_fps<<<16, 256, 0, stream>>>(NX1, FPS2, 512, 128);
  k_gather_nx<<<cdiv((size_t)16 * 128 * 3, 256), 256, 0, stream>>>(NX1, FPS2, NX2, 512, 128);
  k_ball_query<<<cdiv((size_t)16 * 128, 8), 256, 0, stream>>>(NX1, NX2, IDX2, 512, 128, 64, 0.16f);
  k_build_features<<<cdiv((size_t)16 * 128 * 64 * 132, 256), 256, 0, stream>>>(
      NX1, L1P, NX2, IDX2, BUF0, 512, 128, 64, 128, 132);
  const int M2 = 16 * 128 * 64;
  run_layer(BUF0, BUF1, 3, M2, 131, 128);
  run_layer(BUF1, BUF0, 4, M2, 128, 128);
  run_layer(BUF0, BUF1, 5, M2, 128, 256);
  k_max_k<<<cdiv((size_t)16 * 128 * 256, 256), 256, 0, stream>>>(BUF1, L2P, 16 * 128, 64, 256);

  // -------- stage 3: group_all, MLP 259->256->512->1024, max over 128 ------
  k_build_concat3<<<cdiv((size_t)2048 * 260, 256), 256, 0, stream>>>(NX2, L2P, BUF0);
  const int M3 = 2048;
  run_layer(BUF0, BUF1, 6, M3, 259, 256);
  run_layer(BUF1, BUF0, 7, M3, 256, 512);
  run_layer(BUF0, BUF1, 8, M3, 512, 1024);

  float* outF = (float*)d_out;
  k_zero<<<1, 64, 0, stream>>>(outF, 48);                       // l3_xyz == zeros
  k_max_k<<<cdiv((size_t)16 * 1024, 256), 256, 0, stream>>>(BUF1, outF + 48, 16, 128, 1024);
}